// SchNetInteraction_71408126263664
// MI455X (gfx1250) — compile-verified
//
#include <hip/hip_runtime.h>
#include <hip/hip_bf16.h>

#define N_ATOMS 50000
#define N_EDGES 1600000
#define NAB 128
#define NRBF 20
#define NF 128
#define TPAD (NF + 4)   // padded row stride for LDS T staging (bank-conflict break)

typedef __attribute__((ext_vector_type(2))) float v2f;
typedef __attribute__((ext_vector_type(8))) float v8f;

__device__ __forceinline__ v8f wmma_f32x4(v2f a, v2f b, v8f c) {
    // V_WMMA_F32_16X16X4_F32 : D = A(16x4) * B(4x16) + C(16x16), fp32
    return __builtin_amdgcn_wmma_f32_16x16x4_f32(
        /*neg_a=*/false, a, /*neg_b=*/false, b,
        /*c_mod=*/(short)0, c, /*reuse_a=*/false, /*reuse_b=*/false);
}

__device__ __forceinline__ float sspf(float x) {
    // shifted softplus: log(1+exp(x)) - log(2), numerically stable
    return fmaxf(x, 0.f) + log1pf(expf(-fabsf(x))) - 0.69314718055994530942f;
}

// Stage a [KROWS x 128] row-major weight matrix into LDS in K-pair-interleaved
// layout: sW2[(p*128 + col)*2 + j] = W[(2p + j)*128 + col].
// A B-fragment (kk, col) is then ONE 8-byte-aligned ds_load_b64 at
// float offset 2*((kk/2)*128 + col), landing in an even VGPR pair.
__device__ __forceinline__ void stage_w_pairs(const float* __restrict__ W,
                                              float* __restrict__ sW2,
                                              int krows, int tid) {
    const int npairs = (krows / 2) * NF;
    for (int i = tid; i < npairs; i += 256) {
        const int p   = i >> 7;       // k-pair index
        const int col = i & (NF - 1);
        sW2[2 * i + 0] = W[(2 * p)     * NF + col];
        sW2[2 * i + 1] = W[(2 * p + 1) * NF + col];
    }
}

__device__ __forceinline__ v2f ld_pair_lds(const float* __restrict__ sW2,
                                           int kk, int col) {
    return *(const v2f*)(sW2 + (((kk >> 1) * NF + col) << 1));
}

// ---------------------------------------------------------------------------
// zero-fill (float4 vectorized)
// ---------------------------------------------------------------------------
__global__ void zero_f32(float* __restrict__ p, int n4) {
    int i = blockIdx.x * blockDim.x + threadIdx.x;
    if (i < n4) ((float4*)p)[i] = make_float4(0.f, 0.f, 0.f, 0.f);
}

// ---------------------------------------------------------------------------
// out[M,128] = act(A[M,128] @ W[128,128] + bias)   (bias optional, act optional)
// One wave -> 16-row tile x all 128 cols (8 col tiles), K=128 via f32 WMMA.
// W (pair-interleaved) + bias staged in LDS, shared by 8 waves.
// Rows are CLAMPED (not exec-masked) for the A loads; garbage rows only
// pollute garbage D rows which are never stored.
// ---------------------------------------------------------------------------
__global__ __launch_bounds__(256) void gemm128_wmma(
    const float* __restrict__ A, const float* __restrict__ W,
    const float* __restrict__ bias, float* __restrict__ out,
    int M, int apply_ssp)
{
    extern __shared__ float smem[];
    float* sW2 = smem;            // 128*128 (pair-interleaved)
    float* sB  = smem + NF * NF;  // 128

    const int tid = threadIdx.x;
    stage_w_pairs(W, sW2, NF, tid);
    if (tid < NF) sB[tid] = bias ? bias[tid] : 0.f;
    __syncthreads();

    const int lane = tid & 31;
    const int wave = tid >> 5;
    const int m    = lane & 15;  // row (A) or col (B/C) within tile
    const int hi   = lane >> 4;  // half-wave selector
    const int r0   = blockIdx.x * 128 + wave * 16;

    v8f acc[8];
    const v8f z8 = {0.f, 0.f, 0.f, 0.f, 0.f, 0.f, 0.f, 0.f};
    #pragma unroll
    for (int ct = 0; ct < 8; ++ct) acc[ct] = z8;

    const int rowc = min(r0 + m, M - 1);              // clamp, no exec masking
    const float* arow = A + (size_t)rowc * NF;

    #pragma unroll 4
    for (int k = 0; k < NF; k += 4) {
        const int kk = k + hi * 2;
        const v2f a = *(const v2f*)(arow + kk);       // one global_load_b64
        #pragma unroll
        for (int ct = 0; ct < 8; ++ct) {
            const v2f b = ld_pair_lds(sW2, kk, ct * 16 + m);  // one ds_load_b64
            acc[ct] = wmma_f32x4(a, b, acc[ct]);
        }
    }

    #pragma unroll
    for (int ct = 0; ct < 8; ++ct) {
        const int col = ct * 16 + m;
        #pragma unroll
        for (int v = 0; v < 8; ++v) {
            const int r = r0 + v + hi * 8;
            if (r < M) {
                float val = acc[ct][v] + sB[col];
                if (apply_ssp) val = sspf(val);
                out[(size_t)r * NF + col] = val;
            }
        }
    }
}

// ---------------------------------------------------------------------------
// Fused edge pipeline, per 16-edge tile (one wave):
//   T   = ssp(f_ij @ W_f1 + b_f1)            [16 x 128], K=20  (WMMA)
//   Wij = (T @ W_f2 + b_f2) * rcut           [16 x 128], K=128 (WMMA)
//   y[idx_i] += h[idx_j] * Wij               (gather + reduced atomic scatter)
// W_f1/W_f2 pair-interleaved in LDS, shared by the block; T round-trips
// through per-wave padded LDS to convert C-layout -> A-layout.
// idx_i is sorted -> per-lane run-length reduction over the 8 register rows
// before the atomic scatter (cuts atomic traffic ~4-8x).
// ---------------------------------------------------------------------------
__global__ __launch_bounds__(256) void edge_cfconv_wmma(
    const float* __restrict__ f_ij, const float* __restrict__ rcut,
    const int* __restrict__ idx_i, const int* __restrict__ idx_j,
    const float* __restrict__ Wf1, const float* __restrict__ bf1,
    const float* __restrict__ Wf2, const float* __restrict__ bf2,
    const float* __restrict__ h, float* __restrict__ y)
{
    extern __shared__ float smem[];
    float* sWf2 = smem;                    // 128*128 = 16384 (pair-interleaved)
    float* sWf1 = sWf2 + NF * NF;          // 20*128  = 2560  (pair-interleaved)
    float* sB1  = sWf1 + NRBF * NF;        // 128
    float* sB2  = sB1 + NF;                // 128
    float* sT   = sB2 + NF;                // 8 waves * 16 * TPAD

    const int tid = threadIdx.x;
    stage_w_pairs(Wf2, sWf2, NF, tid);
    stage_w_pairs(Wf1, sWf1, NRBF, tid);
    if (tid < NF) { sB1[tid] = bf1[tid]; sB2[tid] = bf2[tid]; }
    __syncthreads();

    const int lane = tid & 31;
    const int wave = tid >> 5;
    const int m    = lane & 15;
    const int hi   = lane >> 4;
    const int eb   = blockIdx.x * 128 + wave * 16;  // N_EDGES % 128 == 0

    float* tw = sT + wave * 16 * TPAD;
    const v8f z8 = {0.f, 0.f, 0.f, 0.f, 0.f, 0.f, 0.f, 0.f};

    // ---- stage 1: T = ssp(f_ij @ W_f1 + b_f1), K = 20 ----
    v8f acc[8];
    #pragma unroll
    for (int ct = 0; ct < 8; ++ct) acc[ct] = z8;

    const float* frow = f_ij + (size_t)(eb + m) * NRBF;
    #pragma unroll
    for (int k = 0; k < NRBF; k += 4) {
        const int kk = k + hi * 2;
        const v2f a = *(const v2f*)(frow + kk);       // 8B-aligned b64 load
        #pragma unroll
        for (int ct = 0; ct < 8; ++ct) {
            const v2f b = ld_pair_lds(sWf1, kk, ct * 16 + m);
            acc[ct] = wmma_f32x4(a, b, acc[ct]);
        }
    }

    // C-layout -> LDS (row-major, padded) with bias + ssp applied
    #pragma unroll
    for (int ct = 0; ct < 8; ++ct) {
        const int col = ct * 16 + m;
        #pragma unroll
        for (int v = 0; v < 8; ++v) {
            const int row = v + hi * 8;
            tw[row * TPAD + col] = sspf(acc[ct][v] + sB1[col]);
        }
    }

    // ---- stage 2: Wij = T @ W_f2 + b_f2, K = 128 ----
    v8f acc2[8];
    #pragma unroll
    for (int ct = 0; ct < 8; ++ct) acc2[ct] = z8;

    const float* trow = tw + m * TPAD;                 // 8B-aligned (TPAD even)
    #pragma unroll 4
    for (int k = 0; k < NF; k += 4) {
        const int kk = k + hi * 2;
        const v2f a = *(const v2f*)(trow + kk);        // one ds_load_b64
        #pragma unroll
        for (int ct = 0; ct < 8; ++ct) {
            const v2f b = ld_pair_lds(sWf2, kk, ct * 16 + m);
            acc2[ct] = wmma_f32x4(a, b, acc2[ct]);
        }
    }

    // ---- epilogue: scale by rcut, gather h[idx_j], run-length-reduce rows
    // ---- sharing the same idx_i, then atomic scatter-add into y[idx_i].
    float red[8];
    #pragma unroll
    for (int ct = 0; ct < 8; ++ct) red[ct] = 0.f;
    int cur_i = idx_i[eb + hi * 8];   // target atom of this lane's first row

    #pragma unroll
    for (int v = 0; v < 8; ++v) {
        const int   e  = eb + v + hi * 8;
        const float rc = rcut[e];
        const int   ji = idx_j[e];
        const int   ii = idx_i[e];
        if (ii != cur_i) {            // flush completed run (rare: ~32 edges/atom)
            float* yrow = y + (size_t)cur_i * NF + m;
            #pragma unroll
            for (int ct = 0; ct < 8; ++ct) {
                atomicAdd(yrow + ct * 16, red[ct]);
                red[ct] = 0.f;
            }
            cur_i = ii;
        }
        const float* hrow = h + (size_t)ji * NF + m;   // base + imm offsets
        #pragma unroll
        for (int ct = 0; ct < 8; ++ct) {
            red[ct] += (acc2[ct][v] + sB2[ct * 16 + m]) * rc * hrow[ct * 16];
        }
    }
    {
        float* yrow = y + (size_t)cur_i * NF + m;      // final flush
        #pragma unroll
        for (int ct = 0; ct < 8; ++ct) atomicAdd(yrow + ct * 16, red[ct]);
    }
}

// ---------------------------------------------------------------------------
extern "C" void kernel_launch(void* const* d_in, const int* in_sizes, int n_in,
                              void* d_out, int out_size, void* d_ws, size_t ws_size,
                              hipStream_t stream) {
    const float* x      = (const float*)d_in[0];
    const float* f_ij   = (const float*)d_in[1];
    const float* rcut   = (const float*)d_in[2];
    const int*   idx_i  = (const int*)d_in[3];
    const int*   idx_j  = (const int*)d_in[4];
    const float* W_in2f = (const float*)d_in[5];
    const float* W_f1   = (const float*)d_in[6];
    const float* b_f1   = (const float*)d_in[7];
    const float* W_f2   = (const float*)d_in[8];
    const float* b_f2   = (const float*)d_in[9];
    const float* W_o1   = (const float*)d_in[10];
    const float* b_o1   = (const float*)d_in[11];
    const float* W_o2   = (const float*)d_in[12];
    const float* b_o2   = (const float*)d_in[13];
    float* out = (float*)d_out;

    float* h = (float*)d_ws;                  // [N_ATOMS, NF], reused for T later
    float* y = h + (size_t)N_ATOMS * NF;      // [N_ATOMS, NF] scatter accumulator

    const size_t gemm_lds = (size_t)(NF * NF + NF) * sizeof(float);
    const size_t edge_lds = (size_t)(NF * NF + NRBF * NF + 2 * NF + 8 * 16 * TPAD) * sizeof(float);

    // y = 0 (re-done every call; harness poisons workspace)
    {
        const int n4 = (N_ATOMS * NF) / 4;
        zero_f32<<<(n4 + 255) / 256, 256, 0, stream>>>(y, n4);
    }
    // h = x @ W_in2f
    gemm128_wmma<<<(N_ATOMS + 127) / 128, 256, gemm_lds, stream>>>(
        x, W_in2f, nullptr, h, N_ATOMS, 0);
    // fused edge filter + cfconv scatter
    edge_cfconv_wmma<<<N_EDGES / 128, 256, edge_lds, stream>>>(
        f_ij, rcut, idx_i, idx_j, W_f1, b_f1, W_f2, b_f2, h, y);
    // t = ssp(y @ W_o1 + b_o1)   (reuse h buffer)
    gemm128_wmma<<<(N_ATOMS + 127) / 128, 256, gemm_lds, stream>>>(
        y, W_o1, b_o1, h, N_ATOMS, 1);
    // out = t @ W_o2 + b_o2
    gemm128_wmma<<<(N_ATOMS + 127) / 128, 256, gemm_lds, stream>>>(
        h, W_o2, b_o2, out, N_ATOMS, 0);
}